// TwoStageSelector_52999896433203
// MI455X (gfx1250) — compile-verified
//
#include <hip/hip_runtime.h>
#include <hip/hip_bf16.h>

#define NUM_BAGS 4096
#define BAG 16
#define D 1024
#define NC_MULTI 53
#define NC_BIN 2

typedef float v2f __attribute__((ext_vector_type(2)));
typedef float v8f __attribute__((ext_vector_type(8)));

// ---------------------------------------------------------------------------
// Prep: c[k][d] = att[k][d] * rel[k][d]  (tiny)
// ---------------------------------------------------------------------------
__global__ void ewise_mul_kernel(const float* __restrict__ a,
                                 const float* __restrict__ b,
                                 float* __restrict__ c, int n) {
  int i = blockIdx.x * blockDim.x + threadIdx.x;
  if (i < n) c[i] = a[i] * b[i];
}

// ---------------------------------------------------------------------------
// One block = one bag of 16 sentences. Single pass over x (the 256 MB stream):
// bag held in registers (float4 per sentence per thread), logits via wave32
// shuffle reduce, softmax in-block, weighted sum -> repre[bag][0..1023].
// ---------------------------------------------------------------------------
__global__ __launch_bounds__(256) void bag_attention_kernel(
    const float* __restrict__ x, const float* __restrict__ ctab,
    const int* __restrict__ query, float* __restrict__ repre) {
  const int b = blockIdx.x;
  const int tid = threadIdx.x;

  __shared__ int qs[BAG];
  __shared__ float red[BAG * 8];
  __shared__ float lg[BAG];

  if (tid < BAG) qs[tid] = query[b * BAG + tid];
  __syncthreads();

  float4 xr[BAG];
  float part[BAG];
#pragma unroll
  for (int s = 0; s < BAG; ++s) {
    const float4 xv = ((const float4*)(x + (size_t)(b * BAG + s) * D))[tid];
    const float4 cv = ((const float4*)(ctab + (size_t)qs[s] * D))[tid];
    xr[s] = xv;
    part[s] = xv.x * cv.x + xv.y * cv.y + xv.z * cv.z + xv.w * cv.w;
  }

  const int lane = tid & 31;
  const int wv = tid >> 5;
#pragma unroll
  for (int s = 0; s < BAG; ++s) {
    float v = part[s];
    for (int off = 16; off > 0; off >>= 1) v += __shfl_down(v, off, 32);
    if (lane == 0) red[s * 8 + wv] = v;
  }
  __syncthreads();

  if (tid < BAG) {
    float t = 0.f;
#pragma unroll
    for (int w = 0; w < 8; ++w) t += red[tid * 8 + w];
    lg[tid] = t;
  }
  __syncthreads();

  // softmax over the 16 bag logits (computed redundantly per thread; cheap)
  float m = -3.402823466e38f;
#pragma unroll
  for (int s = 0; s < BAG; ++s) m = fmaxf(m, lg[s]);
  float e[BAG];
  float sum = 0.f;
#pragma unroll
  for (int s = 0; s < BAG; ++s) { e[s] = expf(lg[s] - m); sum += e[s]; }
  const float inv = 1.0f / sum;

  float4 acc = {0.f, 0.f, 0.f, 0.f};
#pragma unroll
  for (int s = 0; s < BAG; ++s) {
    const float w = e[s] * inv;
    acc.x += w * xr[s].x;
    acc.y += w * xr[s].y;
    acc.z += w * xr[s].z;
    acc.w += w * xr[s].w;
  }
  ((float4*)(repre + (size_t)b * D))[tid] = acc;
}

// ---------------------------------------------------------------------------
// Final projection via V_WMMA_F32_16X16X4_F32:
//   out[16 bags x 16 classes tile] = repre[16x1024] @ rel^T[1024x16] + bias
// One wave (32 threads) per output tile; K-loop of 256 x k=4.
// ISA layouts: A 16x4 -> lanes 0-15 hold K={0,1}, lanes 16-31 hold K={2,3};
// B 4x16 mirrored (lane = N column); C/D: M = r + 8*(lane>=16), N = lane&15.
// ---------------------------------------------------------------------------
__global__ __launch_bounds__(32) void bag_logits_wmma_kernel(
    const float* __restrict__ repre, const float* __restrict__ rel,
    const float* __restrict__ bias, int num_classes, float* __restrict__ out) {
  const int bag0 = blockIdx.x * 16;
  const int n0 = blockIdx.y * 16;
  const int lane = threadIdx.x;
  const int half = lane >> 4;   // which K pair this lane carries
  const int l16 = lane & 15;

  const int cls = n0 + l16;
  const bool cvalid = cls < num_classes;

  const float* arow = repre + (size_t)(bag0 + l16) * D;       // A row (bag)
  const float* brow = rel + (size_t)(cvalid ? cls : 0) * D;   // B col (class)

  v8f acc = {0.f, 0.f, 0.f, 0.f, 0.f, 0.f, 0.f, 0.f};
#pragma unroll 4
  for (int k0 = 0; k0 < D; k0 += 4) {
    const int kk = k0 + half * 2;
    v2f a = *(const v2f*)(arow + kk);
    v2f bb = *(const v2f*)(brow + kk);
    if (!cvalid) { bb.x = 0.f; bb.y = 0.f; }
    acc = __builtin_amdgcn_wmma_f32_16x16x4_f32(
        /*neg_a=*/false, a, /*neg_b=*/false, bb,
        /*c_mod=*/(short)0, acc, /*reuse_a=*/false, /*reuse_b=*/false);
  }

  if (cvalid) {
    const float bv = bias[cls];
#pragma unroll
    for (int r = 0; r < 8; ++r) {
      const int bag = bag0 + r + half * 8;
      out[(size_t)bag * num_classes + cls] = acc[r] + bv;
    }
  }
}

// ---------------------------------------------------------------------------
extern "C" void kernel_launch(void* const* d_in, const int* in_sizes, int n_in,
                              void* d_out, int out_size, void* d_ws, size_t ws_size,
                              hipStream_t stream) {
  const float* soc_x      = (const float*)d_in[0];
  const float* context_x  = (const float*)d_in[1];
  const float* multi_rel  = (const float*)d_in[2];
  const float* multi_att  = (const float*)d_in[3];
  const float* multi_bias = (const float*)d_in[4];
  const float* bin_rel    = (const float*)d_in[5];
  const float* bin_att    = (const float*)d_in[6];
  const float* bin_bias   = (const float*)d_in[7];
  const int* att_q        = (const int*)d_in[8];
  const int* bin_q        = (const int*)d_in[9];
  float* out = (float*)d_out;

  float* ws = (float*)d_ws;
  float* c_multi   = ws;                            // 53*1024
  float* c_bin     = c_multi + NC_MULTI * D;        // 2*1024
  float* repre_soc = c_bin + NC_BIN * D;            // 4096*1024
  float* repre_ctx = repre_soc + (size_t)NUM_BAGS * D;

  ewise_mul_kernel<<<(NC_MULTI * D + 255) / 256, 256, 0, stream>>>(
      multi_att, multi_rel, c_multi, NC_MULTI * D);
  ewise_mul_kernel<<<(NC_BIN * D + 255) / 256, 256, 0, stream>>>(
      bin_att, bin_rel, c_bin, NC_BIN * D);

  bag_attention_kernel<<<NUM_BAGS, 256, 0, stream>>>(soc_x, c_multi, att_q, repre_soc);
  bag_attention_kernel<<<NUM_BAGS, 256, 0, stream>>>(context_x, c_bin, bin_q, repre_ctx);

  bag_logits_wmma_kernel<<<dim3(NUM_BAGS / 16, 4), 32, 0, stream>>>(
      repre_soc, multi_rel, multi_bias, NC_MULTI, out);
  bag_logits_wmma_kernel<<<dim3(NUM_BAGS / 16, 1), 32, 0, stream>>>(
      repre_ctx, bin_rel, bin_bias, NC_BIN, out + (size_t)NUM_BAGS * NC_MULTI);
}